// EquivariantLayer_42099269436238
// MI455X (gfx1250) — compile-verified
//
#include <hip/hip_runtime.h>

#define N_NODES 100000
#define N_EDGES 800000
#define DIMF    160          // 64 scalar + 32*3 vector channels

typedef float v2f __attribute__((ext_vector_type(2)));
typedef float v8f __attribute__((ext_vector_type(8)));

// D(16x16,f32) += A(16x4,f32) * B(4x16,f32); exact f32 path (matches reference precision).
#define WMMA_F32_K4(ACC, A, B) \
  (ACC) = __builtin_amdgcn_wmma_f32_16x16x4_f32(false, (A), false, (B), (short)0, (ACC), false, false)

#define V8F_ZERO {0.f,0.f,0.f,0.f,0.f,0.f,0.f,0.f}

#define EDGE_TILES_PER_BLOCK 8     // 6250 blocks * 8 tiles * 16 edges = 800000
#define NODE_TILES_PER_BLOCK 5     // 1250 blocks * 5 tiles * 16 nodes = 100000

__device__ __forceinline__ float sigmoidf(float t) { return 1.0f / (1.0f + __expf(-t)); }

__global__ void zero_kernel(float* __restrict__ p, int n) {
  int i = blockIdx.x * blockDim.x + threadIdx.x;
  if (i < n) p[i] = 0.0f;
}

// ---------------------------------------------------------------------------
// Edge kernel: one wave (32 lanes) per block; each block packs all weights
// into LDS in WMMA B-fragment order once, then processes 8 tiles of 16 edges.
// Hot loop: 1 ds_load_b64 per v_wmma_f32_16x16x4_f32.
// ---------------------------------------------------------------------------
__global__ __launch_bounds__(32) void edge_kernel(
    const float*     __restrict__ x,
    const long long* __restrict__ edge_index,   // [2, N_EDGES] int64
    const float*     __restrict__ edge_feat,    // [N_EDGES, 4]
    const float*     __restrict__ edge_scalars, // [N_EDGES, 16]
    const float*     __restrict__ W_ss,         // [64,64] (u,o)
    const float*     __restrict__ W_sv,         // [64,32] (u,v)
    const float*     __restrict__ W_vs,         // [32,32] (u,v)
    const float*     __restrict__ W_vv,         // [32,64] (u,o)
    const float*     __restrict__ W_rad,        // [160,16] (j,t)
    const float*     __restrict__ b_rad,        // [160]
    float*           __restrict__ agg)          // [N_NODES, 160]
{
  // Weight fragments, packed so lane `l` reads its v2f with one ds_load_b64.
  __shared__ v2f  sWss[4 * 16 * 32];   // 16 KB : W_ss  (ot, kk)
  __shared__ v2f  sWvv[4 *  8 * 32];   //  8 KB : W_vv  (ot, kk)
  __shared__ v2f  sWsv[2 * 16 * 32];   //  8 KB : W_sv  (vt, kk)
  __shared__ v2f  sWvs[2 *  8 * 32];   //  4 KB : W_vs  (vt, kk)
  __shared__ v2f  sGs [4 *  4 * 32];   //  4 KB : W_rad cols 0..63      (ot, kk)
  __shared__ v2f  sGv [6 *  4 * 32];   //  6 KB : W_rad cols 64+3v+d    (vt,d,kk)
  __shared__ float sBr[DIMF];          // b_rad
  // Per-tile staging.
  __shared__ float sX[16][DIMF];       // gathered x[src] rows
  __shared__ float sEF[16][4];         // es, ev0, ev1, ev2
  __shared__ float sESC[16][16];       // edge scalars
  __shared__ int   sSrc[16];
  __shared__ int   sDst[16];

  const int lane    = threadIdx.x;
  // Fragment lane mapping (CDNA5 ISA 7.12.2):
  //  A 16x4 f32:  M = lane&15; K slots {kb, kb+1}, kb = 2*(lane>>4)
  //  B  4x16 f32: N = lane&15; same K split
  //  C 16x16 f32: N = lane&15; vgpr r -> M = r + 8*(lane>>4)
  const int m       = lane & 15;
  const int halfsel = lane >> 4;
  const int kb      = halfsel << 1;
  const int mrb     = halfsel << 3;

  // ---- pack weights into fragment order (once per block) -----------------
  for (int nt = 0; nt < 4; ++nt)
    for (int kk = 0; kk < 16; ++kk) {
      const int u = kk * 4 + kb;
      sWss[(nt * 16 + kk) * 32 + lane] = (v2f){ W_ss[u * 64 + nt * 16 + m],
                                                W_ss[(u + 1) * 64 + nt * 16 + m] };
    }
  for (int nt = 0; nt < 4; ++nt)
    for (int kk = 0; kk < 8; ++kk) {
      const int u = kk * 4 + kb;
      sWvv[(nt * 8 + kk) * 32 + lane] = (v2f){ W_vv[u * 64 + nt * 16 + m],
                                               W_vv[(u + 1) * 64 + nt * 16 + m] };
    }
  for (int vt = 0; vt < 2; ++vt)
    for (int kk = 0; kk < 16; ++kk) {
      const int u = kk * 4 + kb;
      sWsv[(vt * 16 + kk) * 32 + lane] = (v2f){ W_sv[u * 32 + vt * 16 + m],
                                                W_sv[(u + 1) * 32 + vt * 16 + m] };
    }
  for (int vt = 0; vt < 2; ++vt)
    for (int kk = 0; kk < 8; ++kk) {
      const int u = kk * 4 + kb;
      sWvs[(vt * 8 + kk) * 32 + lane] = (v2f){ W_vs[u * 32 + vt * 16 + m],
                                               W_vs[(u + 1) * 32 + vt * 16 + m] };
    }
  for (int ot = 0; ot < 4; ++ot)
    for (int kk = 0; kk < 4; ++kk) {
      const int t = kk * 4 + kb, col = ot * 16 + m;
      sGs[(ot * 4 + kk) * 32 + lane] = (v2f){ W_rad[col * 16 + t], W_rad[col * 16 + t + 1] };
    }
  for (int vt = 0; vt < 2; ++vt)
    for (int d = 0; d < 3; ++d)
      for (int kk = 0; kk < 4; ++kk) {
        const int t = kk * 4 + kb, col = 64 + (vt * 16 + m) * 3 + d;
        sGv[(((vt * 3 + d) * 4) + kk) * 32 + lane] =
            (v2f){ W_rad[col * 16 + t], W_rad[col * 16 + t + 1] };
      }
  for (int j = lane; j < DIMF; j += 32) sBr[j] = b_rad[j];
  __syncthreads();

  const float cSS = 0.0883883476483f;   // ISQ2 / sqrt(64)
  const float cVV = 0.0721687836487f;   // ISQ2 * ISQ3 / sqrt(32)  (== cVS)
  const float cSV = 0.0510310363080f;   // ISQ2 * ISQ3 / sqrt(64)

  const int tile0 = blockIdx.x * EDGE_TILES_PER_BLOCK;
  for (int it = 0; it < EDGE_TILES_PER_BLOCK; ++it) {
    const int e0 = (tile0 + it) * 16;
    if (it + 1 < EDGE_TILES_PER_BLOCK) {   // prefetch next tile's edge streams
      const int en = e0 + 16;
      __builtin_prefetch(edge_scalars + (size_t)en * 16, 0, 1);
      __builtin_prefetch(edge_feat + (size_t)en * 4, 0, 1);
    }
    // ---- stage tile ------------------------------------------------------
    if (lane < 16) {
      const int e = e0 + lane;
      sSrc[lane] = (int)edge_index[e];
      sDst[lane] = (int)edge_index[N_EDGES + e];
      float4 ef = *(const float4*)(edge_feat + (size_t)e * 4);
      sEF[lane][0] = ef.x; sEF[lane][1] = ef.y; sEF[lane][2] = ef.z; sEF[lane][3] = ef.w;
    }
    {
      const float4* esrc = (const float4*)(edge_scalars + (size_t)e0 * 16);
      float4* edst = (float4*)(&sESC[0][0]);
      edst[lane]      = esrc[lane];
      edst[lane + 32] = esrc[lane + 32];
    }
    __syncthreads();
    for (int r = 0; r < 16; ++r) {
      const float4* xr = (const float4*)(x + (size_t)sSrc[r] * DIMF);
      float4* dr = (float4*)(&sX[r][0]);
      dr[lane] = xr[lane];
      if (lane < 8) dr[lane + 32] = xr[lane + 32];
    }
    __syncthreads();

    const float esm = sEF[m][0];
    const float ev0 = sEF[m][1], ev1 = sEF[m][2], ev2 = sEF[m][3];
    const float esS = esm * cSS;
    const float esV = esm * cVV;

    // ---- out_s: 4 column tiles ------------------------------------------
    for (int ot = 0; ot < 4; ++ot) {
      v8f acc  = V8F_ZERO;
      v8f gacc = V8F_ZERO;
      #pragma unroll
      for (int kk = 0; kk < 16; ++kk) {         // (xs*es) @ W_ss, K=64
        const int u = kk * 4 + kb;
        v2f a; a.x = sX[m][u] * esS; a.y = sX[m][u + 1] * esS;
        WMMA_F32_K4(acc, a, sWss[(ot * 16 + kk) * 32 + lane]);
      }
      #pragma unroll
      for (int kk = 0; kk < 8; ++kk) {          // (xv . ev) @ W_vv, K=32
        const int u = kk * 4 + kb;
        v2f a;
        a.x = (sX[m][64 + u * 3] * ev0 + sX[m][64 + u * 3 + 1] * ev1 +
               sX[m][64 + u * 3 + 2] * ev2) * cVV;
        a.y = (sX[m][64 + (u + 1) * 3] * ev0 + sX[m][64 + (u + 1) * 3 + 1] * ev1 +
               sX[m][64 + (u + 1) * 3 + 2] * ev2) * cVV;
        WMMA_F32_K4(acc, a, sWvv[(ot * 8 + kk) * 32 + lane]);
      }
      #pragma unroll
      for (int kk = 0; kk < 4; ++kk) {          // gate tile, K=16
        v2f a = *(const v2f*)&sESC[m][kk * 4 + kb];
        WMMA_F32_K4(gacc, a, sGs[(ot * 4 + kk) * 32 + lane]);
      }
      const int col = ot * 16 + m;
      const float br = sBr[col];
      #pragma unroll
      for (int r = 0; r < 8; ++r) {
        const float g = sigmoidf(gacc[r] + br);
        atomicAdd(&agg[(size_t)sDst[r + mrb] * DIMF + col], acc[r] * g);
      }
    }

    // ---- out_v: 2 v-tiles x 3 components --------------------------------
    for (int vt = 0; vt < 2; ++vt) {
      v8f pacc = V8F_ZERO;                      // P = xs @ W_sv (reused for d)
      #pragma unroll
      for (int kk = 0; kk < 16; ++kk) {
        const int u = kk * 4 + kb;
        v2f a; a.x = sX[m][u] * cSV; a.y = sX[m][u + 1] * cSV;
        WMMA_F32_K4(pacc, a, sWsv[(vt * 16 + kk) * 32 + lane]);
      }
      for (int d = 0; d < 3; ++d) {
        v8f qacc = V8F_ZERO;
        v8f gacc = V8F_ZERO;
        #pragma unroll
        for (int kk = 0; kk < 8; ++kk) {        // (xv_d * es) @ W_vs, K=32
          const int u = kk * 4 + kb;
          v2f a;
          a.x = sX[m][64 + u * 3 + d] * esV;
          a.y = sX[m][64 + (u + 1) * 3 + d] * esV;
          WMMA_F32_K4(qacc, a, sWvs[(vt * 8 + kk) * 32 + lane]);
        }
        #pragma unroll
        for (int kk = 0; kk < 4; ++kk) {        // gate tile at cols 64+3v+d
          v2f a = *(const v2f*)&sESC[m][kk * 4 + kb];
          WMMA_F32_K4(gacc, a, sGv[(((vt * 3 + d) * 4) + kk) * 32 + lane]);
        }
        const int col = 64 + (vt * 16 + m) * 3 + d;
        const float br = sBr[col];
        #pragma unroll
        for (int r = 0; r < 8; ++r) {
          const int mr = r + mrb;
          const float evd = sEF[mr][1 + d];
          const float g   = sigmoidf(gacc[r] + br);
          atomicAdd(&agg[(size_t)sDst[mr] * DIMF + col], (evd * pacc[r] + qacc[r]) * g);
        }
      }
    }
    __syncthreads();   // protect sX/sESC against next tile's restaging
  }
}

// ---------------------------------------------------------------------------
// Node kernel: one wave per block, L_s/L_v fragment-packed in LDS once,
// 5 tiles of 16 nodes per block. h = x + agg; apply L_s, L_v.
// ---------------------------------------------------------------------------
__global__ __launch_bounds__(32) void node_kernel(
    const float* __restrict__ x,
    const float* __restrict__ agg,
    const float* __restrict__ L_s,   // [64,64]
    const float* __restrict__ L_v,   // [32,32]
    float*       __restrict__ out)   // [N_NODES, 160]
{
  __shared__ v2f  sLs[4 * 16 * 32];  // 16 KB
  __shared__ v2f  sLv[2 *  8 * 32];  //  4 KB
  __shared__ float sH[16][DIMF];

  const int lane    = threadIdx.x;
  const int m       = lane & 15;
  const int halfsel = lane >> 4;
  const int kb      = halfsel << 1;
  const int mrb     = halfsel << 3;

  for (int nt = 0; nt < 4; ++nt)
    for (int kk = 0; kk < 16; ++kk) {
      const int u = kk * 4 + kb;
      sLs[(nt * 16 + kk) * 32 + lane] = (v2f){ L_s[u * 64 + nt * 16 + m],
                                               L_s[(u + 1) * 64 + nt * 16 + m] };
    }
  for (int vt = 0; vt < 2; ++vt)
    for (int kk = 0; kk < 8; ++kk) {
      const int u = kk * 4 + kb;
      sLv[(vt * 8 + kk) * 32 + lane] = (v2f){ L_v[u * 32 + vt * 16 + m],
                                              L_v[(u + 1) * 32 + vt * 16 + m] };
    }
  __syncthreads();

  const float cS = 0.125f;            // 1/sqrt(64)
  const float cV = 0.17677669529f;    // 1/sqrt(32)

  const int tileBase = blockIdx.x * NODE_TILES_PER_BLOCK;
  for (int it = 0; it < NODE_TILES_PER_BLOCK; ++it) {
    const int n0 = (tileBase + it) * 16;
    for (int r = 0; r < 16; ++r) {
      const float4* xr = (const float4*)(x   + (size_t)(n0 + r) * DIMF);
      const float4* ar = (const float4*)(agg + (size_t)(n0 + r) * DIMF);
      float4* hr = (float4*)(&sH[r][0]);
      float4 a = xr[lane], b = ar[lane];
      hr[lane] = make_float4(a.x + b.x, a.y + b.y, a.z + b.z, a.w + b.w);
      if (lane < 8) {
        float4 a2 = xr[lane + 32], b2 = ar[lane + 32];
        hr[lane + 32] = make_float4(a2.x + b2.x, a2.y + b2.y, a2.z + b2.z, a2.w + b2.w);
      }
    }
    __syncthreads();

    for (int ot = 0; ot < 4; ++ot) {
      v8f acc = V8F_ZERO;
      #pragma unroll
      for (int kk = 0; kk < 16; ++kk) {
        const int u = kk * 4 + kb;
        v2f a; a.x = sH[m][u] * cS; a.y = sH[m][u + 1] * cS;
        WMMA_F32_K4(acc, a, sLs[(ot * 16 + kk) * 32 + lane]);
      }
      #pragma unroll
      for (int r = 0; r < 8; ++r)
        out[(size_t)(n0 + r + mrb) * DIMF + ot * 16 + m] = acc[r];
    }

    for (int vt = 0; vt < 2; ++vt)
      for (int d = 0; d < 3; ++d) {
        v8f acc = V8F_ZERO;
        #pragma unroll
        for (int kk = 0; kk < 8; ++kk) {
          const int u = kk * 4 + kb;
          v2f a; a.x = sH[m][64 + u * 3 + d] * cV; a.y = sH[m][64 + (u + 1) * 3 + d] * cV;
          WMMA_F32_K4(acc, a, sLv[(vt * 8 + kk) * 32 + lane]);
        }
        #pragma unroll
        for (int r = 0; r < 8; ++r)
          out[(size_t)(n0 + r + mrb) * DIMF + 64 + (vt * 16 + m) * 3 + d] = acc[r];
      }
    __syncthreads();   // protect sH before restaging
  }
}

extern "C" void kernel_launch(void* const* d_in, const int* in_sizes, int n_in,
                              void* d_out, int out_size, void* d_ws, size_t ws_size,
                              hipStream_t stream) {
  const float*     x            = (const float*)d_in[0];
  const long long* edge_index   = (const long long*)d_in[1];
  const float*     edge_feat    = (const float*)d_in[2];
  const float*     edge_scalars = (const float*)d_in[3];
  const float*     W_ss         = (const float*)d_in[4];
  const float*     W_sv         = (const float*)d_in[5];
  const float*     W_vs         = (const float*)d_in[6];
  const float*     W_vv         = (const float*)d_in[7];
  const float*     W_rad        = (const float*)d_in[8];
  const float*     b_rad        = (const float*)d_in[9];
  const float*     L_s          = (const float*)d_in[10];
  const float*     L_v          = (const float*)d_in[11];

  float* agg = (float*)d_ws;                        // N_NODES*DIMF f32 = 64 MB scratch
  const int agg_n = N_NODES * DIMF;

  zero_kernel<<<(agg_n + 255) / 256, 256, 0, stream>>>(agg, agg_n);

  const int edge_blocks = N_EDGES / (16 * EDGE_TILES_PER_BLOCK);   // 6250
  edge_kernel<<<edge_blocks, 32, 0, stream>>>(x, edge_index, edge_feat, edge_scalars,
                                              W_ss, W_sv, W_vs, W_vv, W_rad, b_rad, agg);

  const int node_blocks = N_NODES / (16 * NODE_TILES_PER_BLOCK);   // 1250
  node_kernel<<<node_blocks, 32, 0, stream>>>(x, agg, L_s, L_v, (float*)d_out);
}